// LinearQuant_26233660244146
// MI455X (gfx1250) — compile-verified
//
#include <hip/hip_runtime.h>

// Clang extended vector type -> lowers to global_load_b128 / global_store_b128
typedef float v4f __attribute__((ext_vector_type(4)));

__device__ __forceinline__ float quant1(float x) {
    // reference: floor(x/delta + 0.5), delta = 2^-4 -> x*16 is exact,
    // so fmaf(x,16,0.5) == RN(x*16 + 0.5) bit-exactly.
    float r = floorf(fmaf(x, 16.0f, 0.5f));
    r = fmaxf(r, -128.0f);   // -bound
    r = fminf(r,  127.0f);   //  bound - 1
    return r * 0.0625f;      // * delta
}

__global__ __launch_bounds__(256)
void LinearQuant_26233660244146_kernel(const v4f* __restrict__ x,
                                       v4f* __restrict__ out,
                                       long long nvec,
                                       const float* __restrict__ xtail,
                                       float* __restrict__ otail,
                                       int rem) {
    const long long stride = (long long)gridDim.x * (long long)blockDim.x;
    long long i = (long long)blockIdx.x * (long long)blockDim.x + threadIdx.x;

    // Streamed-once data: non-temporal both ways so the 411MB stream does not
    // thrash the 192MB L2. Each wave32 iteration moves 512B/load + 512B/store.
    for (; i < nvec; i += stride) {
        v4f v = __builtin_nontemporal_load(x + i);
        v4f r;
        r.x = quant1(v.x);
        r.y = quant1(v.y);
        r.z = quant1(v.z);
        r.w = quant1(v.w);
        __builtin_nontemporal_store(r, out + i);
    }

    // Scalar tail (n % 4), handled by block 0 only.
    if (blockIdx.x == 0 && (int)threadIdx.x < rem) {
        otail[threadIdx.x] = quant1(xtail[threadIdx.x]);
    }
}

extern "C" void kernel_launch(void* const* d_in, const int* in_sizes, int n_in,
                              void* d_out, int out_size, void* d_ws, size_t ws_size,
                              hipStream_t stream) {
    (void)n_in; (void)d_ws; (void)ws_size; (void)out_size;

    const float* x = (const float*)d_in[0];
    float* out = (float*)d_out;

    const long long n    = (long long)in_sizes[0];   // 51,380,224 for this shape
    const long long nvec = n >> 2;                   // float4 count
    const int rem        = (int)(n & 3LL);

    const int threads = 256;                         // 8 waves (wave32) per block
    // Target ~8 float4 (32 elements) per thread: enough ILP per wave, and
    // thousands of blocks to fill every WGP while keeping launch trivial.
    long long want = (nvec + (long long)threads * 8 - 1) / ((long long)threads * 8);
    if (want < 1) want = 1;
    if (want > 65536) want = 65536;
    const int blocks = (int)want;

    LinearQuant_26233660244146_kernel<<<blocks, threads, 0, stream>>>(
        (const v4f*)x, (v4f*)out, nvec, x + (nvec << 2), out + (nvec << 2), rem);
}